// GAT_HGNNConv_87436944212361
// MI455X (gfx1250) — compile-verified
//
#include <hip/hip_runtime.h>
#include <hip/hip_bf16.h>
#include <math.h>

typedef __attribute__((ext_vector_type(2))) float v2f;
typedef __attribute__((ext_vector_type(8))) float v8f;

#define N_NODES 50000
#define N_EDGES 800000
#define M_HYPER 10000
#define P_PAIRS 200000
#define DIM     64
#define NEG_SLOPE 0.2f

// ---- monotone float<->uint encoding for atomicMax on floats ----
__device__ __forceinline__ unsigned enc_f32(float f) {
    unsigned u = __float_as_uint(f);
    return (u & 0x80000000u) ? ~u : (u | 0x80000000u);
}
__device__ __forceinline__ float dec_f32(unsigned u) {
    return (u & 0x80000000u) ? __uint_as_float(u & 0x7FFFFFFFu)
                             : __uint_as_float(~u);
}

// ---- zero init of accumulator region ----
__global__ void k_zero(float* __restrict__ p, long n) {
    long i = (long)blockIdx.x * blockDim.x + threadIdx.x;
    if (i < n) p[i] = 0.0f;
}

// ---- Xp = X @ W^T + b via V_WMMA_F32_16X16X4_F32, one wave per 16-row tile ----
__global__ void k_gemm_wmma(const float* __restrict__ X, const float* __restrict__ W,
                            const float* __restrict__ bias, float* __restrict__ Xp) {
    int wave = (blockIdx.x * blockDim.x + threadIdx.x) >> 5;
    int lane = threadIdx.x & 31;
    if (wave >= N_NODES / 16) return;           // wave-uniform exit, EXEC stays all-1s
    int row_base = wave * 16;
    int lh  = (lane >= 16) ? 1 : 0;             // lane half selects K sub-pair
    int l15 = lane & 15;

    // A fragment source: row (row_base+l15), k offset lh*2 within each 4-wide k step
    const float* xrow = X + (size_t)(row_base + l15) * DIM + lh * 2;
    // B fragment source: B[k][n] = W[n][k]; lane owns column n = 16*t + l15
    const float* wrow = W + (size_t)l15 * DIM + lh * 2;

    v8f acc[4] = {};
#pragma unroll
    for (int ks = 0; ks < 16; ++ks) {
        v2f a;
        a.x = xrow[ks * 4 + 0];
        a.y = xrow[ks * 4 + 1];
#pragma unroll
        for (int t = 0; t < 4; ++t) {
            v2f b;
            const float* wb = wrow + (size_t)t * 16 * DIM + ks * 4;
            b.x = wb[0];
            b.y = wb[1];
            acc[t] = __builtin_amdgcn_wmma_f32_16x16x4_f32(
                false, a, false, b, (short)0, acc[t], false, false);
        }
    }

    // C/D layout: VGPR v holds M = v + lh*8, N = l15 (per 16-col tile)
#pragma unroll
    for (int t = 0; t < 4; ++t) {
        int c = t * 16 + l15;
        float bv = bias[c];
#pragma unroll
        for (int v = 0; v < 8; ++v) {
            int m = row_base + v + lh * 8;
            Xp[(size_t)m * DIM + c] = acc[t][v] + bv;
        }
    }
}

// ---- per-node attention scalars: s_src = Xp.a_src, s_dst = Xp.a_dst ----
__global__ void k_scalars(const float* __restrict__ Xp,
                          const float* __restrict__ a_src, const float* __restrict__ a_dst,
                          float* __restrict__ s_src, float* __restrict__ s_dst) {
    int n = blockIdx.x * blockDim.x + threadIdx.x;
    if (n >= N_NODES) return;
    const float4* row = (const float4*)(Xp + (size_t)n * DIM);
    const float4* as  = (const float4*)a_src;
    const float4* ad  = (const float4*)a_dst;
    float ss = 0.f, sd = 0.f;
#pragma unroll
    for (int i = 0; i < 16; ++i) {
        float4 x = row[i], u = as[i], v = ad[i];
        ss += x.x * u.x + x.y * u.y + x.z * u.z + x.w * u.w;
        sd += x.x * v.x + x.y * v.y + x.z * v.z + x.w * v.w;
    }
    s_src[n] = ss;
    s_dst[n] = sd;
}

// ---- hypergraph degrees ----
__global__ void k_deg(const int* __restrict__ inc_v, const int* __restrict__ inc_e,
                      float* __restrict__ dv, float* __restrict__ de) {
    int p = blockIdx.x * blockDim.x + threadIdx.x;
    if (p >= P_PAIRS) return;
    atomicAdd(dv + inc_v[p], 1.0f);
    atomicAdd(de + inc_e[p], 1.0f);
}

// ---- edge scores + segment max ----
__global__ void k_score(const float* __restrict__ s_src, const float* __restrict__ s_dst,
                        const int* __restrict__ e_src, const int* __restrict__ e_dst,
                        float* __restrict__ score, unsigned* __restrict__ segmax) {
    int e = blockIdx.x * blockDim.x + threadIdx.x;
    if (e >= N_EDGES) return;
    float x = s_src[e_src[e]] + s_dst[e_dst[e]];
    float sc = (x >= 0.f) ? x : NEG_SLOPE * x;
    score[e] = sc;
    atomicMax(segmax + e_dst[e], enc_f32(sc));
}

// ---- exp(score - max) + segment sum (in-place on score buffer) ----
__global__ void k_exp(const int* __restrict__ e_dst, const unsigned* __restrict__ segmax,
                      float* __restrict__ score, float* __restrict__ denom) {
    int e = blockIdx.x * blockDim.x + threadIdx.x;
    if (e >= N_EDGES) return;
    int d = e_dst[e];
    float m = dec_f32(segmax[d]);
    if (!isfinite(m)) m = 0.f;
    float w = expf(score[e] - m);
    score[e] = w;
    atomicAdd(denom + d, w);
}

// ---- weighted scatter-sum of source features: one wave per edge ----
__global__ void k_scatter_gat(const int* __restrict__ e_src, const int* __restrict__ e_dst,
                              const float* __restrict__ score, const float* __restrict__ denom,
                              const float* __restrict__ Xp, float* __restrict__ Xg) {
    long gid = (long)blockIdx.x * blockDim.x + threadIdx.x;
    long e   = gid >> 5;
    int lane = (int)(gid & 31);
    if (e >= N_EDGES) return;
    int s = e_src[e], d = e_dst[e];
    float coef = score[e] / denom[d];
    float2 xv = *(const float2*)(Xp + (size_t)s * DIM + lane * 2);
    float* out = Xg + (size_t)d * DIM + lane * 2;
    atomicAdd(out + 0, coef * xv.x);
    atomicAdd(out + 1, coef * xv.y);
}

// ---- HGNN: He += dv^{-1/2}[v] * Xp[v], one wave per incidence pair ----
__global__ void k_hg_gather(const int* __restrict__ inc_v, const int* __restrict__ inc_e,
                            const float* __restrict__ Xp, const float* __restrict__ dv,
                            float* __restrict__ He) {
    long gid = (long)blockIdx.x * blockDim.x + threadIdx.x;
    long p   = gid >> 5;
    int lane = (int)(gid & 31);
    if (p >= P_PAIRS) return;
    int v = inc_v[p], e = inc_e[p];
    float dvv = dv[v];
    float s = (dvv > 0.f) ? rsqrtf(dvv) : 0.f;
    float2 xv = *(const float2*)(Xp + (size_t)v * DIM + lane * 2);
    float* out = He + (size_t)e * DIM + lane * 2;
    atomicAdd(out + 0, s * xv.x);
    atomicAdd(out + 1, s * xv.y);
}

// ---- He *= de^{-1} ----
__global__ void k_he_norm(float* __restrict__ He, const float* __restrict__ de) {
    int i = blockIdx.x * blockDim.x + threadIdx.x;
    if (i >= M_HYPER * DIM) return;
    float d = de[i >> 6];
    He[i] *= (d > 0.f) ? (1.f / d) : 0.f;
}

// ---- X_hg += He[e], one wave per pair ----
__global__ void k_hg_scatter(const int* __restrict__ inc_v, const int* __restrict__ inc_e,
                             const float* __restrict__ He, float* __restrict__ Xhg) {
    long gid = (long)blockIdx.x * blockDim.x + threadIdx.x;
    long p   = gid >> 5;
    int lane = (int)(gid & 31);
    if (p >= P_PAIRS) return;
    int v = inc_v[p], e = inc_e[p];
    float2 h = *(const float2*)(He + (size_t)e * DIM + lane * 2);
    float* out = Xhg + (size_t)v * DIM + lane * 2;
    atomicAdd(out + 0, h.x);
    atomicAdd(out + 1, h.y);
}

// ---- out = elu(0.5 * (X_g + X_hg * dv^{-1/2})) ----
__global__ void k_final(const float* __restrict__ Xg, const float* __restrict__ Xhg,
                        const float* __restrict__ dv, float* __restrict__ out) {
    long i = (long)blockIdx.x * blockDim.x + threadIdx.x;
    if (i >= (long)N_NODES * DIM) return;
    float d = dv[i >> 6];
    float s = (d > 0.f) ? rsqrtf(d) : 0.f;
    float x = 0.5f * (Xg[i] + Xhg[i] * s);
    out[i] = (x > 0.f) ? x : expm1f(x);
}

extern "C" void kernel_launch(void* const* d_in, const int* in_sizes, int n_in,
                              void* d_out, int out_size, void* d_ws, size_t ws_size,
                              hipStream_t stream) {
    const float* X     = (const float*)d_in[0];
    const float* W     = (const float*)d_in[1];
    const float* bias  = (const float*)d_in[2];
    const float* a_src = (const float*)d_in[3];
    const float* a_dst = (const float*)d_in[4];
    const int*   e_src = (const int*)d_in[5];
    const int*   e_dst = (const int*)d_in[6];
    const int*   inc_v = (const int*)d_in[7];
    const int*   inc_e = (const int*)d_in[8];
    float* out = (float*)d_out;
    float* ws  = (float*)d_ws;

    size_t off = 0;
    float* Xp    = ws + off; off += (size_t)N_NODES * DIM;
    float* s_src = ws + off; off += N_NODES;
    float* s_dst = ws + off; off += N_NODES;
    size_t zero_base = off;
    unsigned* segmax = (unsigned*)(ws + off); off += N_NODES;
    float* denom = ws + off; off += N_NODES;
    float* dv    = ws + off; off += N_NODES;
    float* de    = ws + off; off += M_HYPER;
    float* Xg    = ws + off; off += (size_t)N_NODES * DIM;
    float* He    = ws + off; off += (size_t)M_HYPER * DIM;
    float* Xhg   = ws + off; off += (size_t)N_NODES * DIM;
    size_t zero_cnt = off - zero_base;
    float* score = ws + off; off += N_EDGES;

    const int B = 256;
    k_zero<<<(int)((zero_cnt + B - 1) / B), B, 0, stream>>>(ws + zero_base, (long)zero_cnt);

    int tiles = N_NODES / 16;                       // 3125, exact
    k_gemm_wmma<<<(tiles * 32 + B - 1) / B, B, 0, stream>>>(X, W, bias, Xp);

    k_scalars<<<(N_NODES + B - 1) / B, B, 0, stream>>>(Xp, a_src, a_dst, s_src, s_dst);
    k_deg<<<(P_PAIRS + B - 1) / B, B, 0, stream>>>(inc_v, inc_e, dv, de);
    k_score<<<(N_EDGES + B - 1) / B, B, 0, stream>>>(s_src, s_dst, e_src, e_dst, score, segmax);
    k_exp<<<(N_EDGES + B - 1) / B, B, 0, stream>>>(e_dst, segmax, score, denom);
    k_scatter_gat<<<(int)(((long)N_EDGES * 32 + B - 1) / B), B, 0, stream>>>(
        e_src, e_dst, score, denom, Xp, Xg);
    k_hg_gather<<<(int)(((long)P_PAIRS * 32 + B - 1) / B), B, 0, stream>>>(
        inc_v, inc_e, Xp, dv, He);
    k_he_norm<<<(M_HYPER * DIM + B - 1) / B, B, 0, stream>>>(He, de);
    k_hg_scatter<<<(int)(((long)P_PAIRS * 32 + B - 1) / B), B, 0, stream>>>(
        inc_v, inc_e, He, Xhg);
    k_final<<<(int)(((long)N_NODES * DIM + B - 1) / B), B, 0, stream>>>(Xg, Xhg, dv, out);
}